// GNN_node_84335977824920
// MI455X (gfx1250) — compile-verified
//
#include <hip/hip_runtime.h>
#include <hip/hip_bf16.h>

#define DIM 128
#define NUM_L 5
#define BOND_V 8
#define LDSA_STRIDE 132  // pad 16x128 A-tile rows to avoid 16-way LDS bank conflicts

typedef __attribute__((ext_vector_type(2))) float v2f;
typedef __attribute__((ext_vector_type(8))) float v8f;

// ---------------------------------------------------------------------------
// AtomEncoder: h[n][d] = sum_f atom_emb[f][x[n][f]][d]
// ---------------------------------------------------------------------------
__global__ __launch_bounds__(128) void atom_encoder_kernel(
    const int* __restrict__ x, const float* __restrict__ atom_emb,
    float* __restrict__ h, int n_nodes) {
  __shared__ int xr[9];
  const int n = blockIdx.x;
  const int d = threadIdx.x;
  if (n >= n_nodes) return;
  if (d < 9) xr[d] = x[n * 9 + d];
  __syncthreads();
  float s = 0.f;
#pragma unroll
  for (int f = 0; f < 9; ++f)
    s += atom_emb[((size_t)f * 128 + xr[f]) * DIM + d];
  h[(size_t)n * DIM + d] = s;
}

// ---------------------------------------------------------------------------
// Degree / norm precompute
// ---------------------------------------------------------------------------
__global__ __launch_bounds__(256) void deg_accum_kernel(
    const int* __restrict__ row, float* __restrict__ deg, int n_edges) {
  int e = blockIdx.x * 256 + threadIdx.x;
  if (e < n_edges) atomicAdd(&deg[row[e]], 1.0f);
}

__global__ __launch_bounds__(256) void deg_finalize_kernel(
    float* __restrict__ deg, float* __restrict__ dinv, int n_nodes) {
  int n = blockIdx.x * 256 + threadIdx.x;
  if (n < n_nodes) {
    float dv = deg[n] + 1.0f;
    deg[n] = dv;
    dinv[n] = rsqrtf(dv);
  }
}

__global__ __launch_bounds__(256) void norm_kernel(
    const int* __restrict__ row, const int* __restrict__ col,
    const float* __restrict__ dinv, float* __restrict__ nrm, int n_edges) {
  int e = blockIdx.x * 256 + threadIdx.x;
  if (e < n_edges) nrm[e] = dinv[row[e]] * dinv[col[e]];
}

// ---------------------------------------------------------------------------
// GEMM: h_lin = h @ W + b   (N x 128) @ (128 x 128), f32 WMMA 16x16x4.
// Block = 256 threads = 8 waves; block tile = 16 rows x 128 cols;
// wave w owns columns [16w, 16w+16). A-tile staged in LDS, B in registers.
// ---------------------------------------------------------------------------
__global__ __launch_bounds__(256) void gemm_wmma_kernel(
    const float* __restrict__ h, const float* __restrict__ W,
    const float* __restrict__ bias, float* __restrict__ out, int n_rows) {
  __shared__ float Atile[16 * LDSA_STRIDE];
  const int t = threadIdx.x;
  const int wave = t >> 5;
  const int lane = t & 31;
  const int row_base = blockIdx.x * 16;

  // Cooperative A-tile load: 16 x 128 floats, 8 floats per thread (2x B128).
  {
    const int r = t >> 4;
    const int c0 = (t & 15) * 8;
    int gr = row_base + r;
    if (gr >= n_rows) gr = n_rows - 1;  // clamp (dup loads, stores guarded)
    const float4* src = (const float4*)(h + (size_t)gr * DIM + c0);
    float4 a0 = src[0];
    float4 a1 = src[1];
    *(float4*)(&Atile[r * LDSA_STRIDE + c0]) = a0;
    *(float4*)(&Atile[r * LDSA_STRIDE + c0 + 4]) = a1;
  }

  // B fragments (ISA 16x16x4 layout): lane holds B[k][j],
  // j = 16*wave + lane%16, k = 4*kt + 2*(lane/16) + v
  const int jcol = wave * 16 + (lane & 15);
  const int khalf = (lane >> 4) * 2;
  v2f Bfrag[32];
#pragma unroll
  for (int kt = 0; kt < 32; ++kt) {
    int k = kt * 4 + khalf;
    Bfrag[kt].x = W[(size_t)k * DIM + jcol];
    Bfrag[kt].y = W[(size_t)(k + 1) * DIM + jcol];
  }

  __syncthreads();

  v8f acc = {};
  const int arow = lane & 15;
#pragma unroll
  for (int kt = 0; kt < 32; ++kt) {
    int k = kt * 4 + khalf;
    v2f a;
    a.x = Atile[arow * LDSA_STRIDE + k];
    a.y = Atile[arow * LDSA_STRIDE + k + 1];
    acc = __builtin_amdgcn_wmma_f32_16x16x4_f32(
        /*neg_a=*/false, a, /*neg_b=*/false, Bfrag[kt],
        /*c_mod=*/(short)0, acc, /*reuse_a=*/false, /*reuse_b=*/false);
  }

  // C/D layout: col = lane%16, row = r + 8*(lane/16)
  const float bv = bias[jcol];
  const int mofs = (lane >> 4) * 8;
#pragma unroll
  for (int r = 0; r < 8; ++r) {
    int m = row_base + mofs + r;
    if (m < n_rows) out[(size_t)m * DIM + jcol] = acc[r] + bv;
  }
}

// ---------------------------------------------------------------------------
// Per-edge message + scatter-add:
//   agg[col] += norm * relu(h_lin[row] + bond_emb_sum)
// bond table (3*8*128 f32 = 12 KB) staged in LDS; 32 threads/edge, float4/thread.
// ---------------------------------------------------------------------------
__global__ __launch_bounds__(256) void message_kernel(
    const int* __restrict__ row, const int* __restrict__ col,
    const int* __restrict__ edge_attr, const float* __restrict__ nrm,
    const float* __restrict__ h_lin, const float* __restrict__ bond_emb_l,
    float* __restrict__ agg, int n_edges) {
  __shared__ float be[3 * BOND_V * DIM];
  for (int i = threadIdx.x; i < 3 * BOND_V * DIM; i += 256)
    be[i] = bond_emb_l[i];
  __syncthreads();

  const int gid = blockIdx.x * 256 + threadIdx.x;
  const int e = gid >> 5;
  if (e >= n_edges) return;
  const int d0 = (gid & 31) * 4;

  const int r = row[e];
  const int c = col[e];
  const float nm = nrm[e];
  const int a0 = edge_attr[e * 3 + 0];
  const int a1 = edge_attr[e * 3 + 1];
  const int a2 = edge_attr[e * 3 + 2];

  const float4 hl = *(const float4*)(h_lin + (size_t)r * DIM + d0);
  const float4 b0 = *(const float4*)(&be[(0 * BOND_V + a0) * DIM + d0]);
  const float4 b1 = *(const float4*)(&be[(1 * BOND_V + a1) * DIM + d0]);
  const float4 b2 = *(const float4*)(&be[(2 * BOND_V + a2) * DIM + d0]);

  float4 m;
  m.x = nm * fmaxf(hl.x + b0.x + b1.x + b2.x, 0.f);
  m.y = nm * fmaxf(hl.y + b0.y + b1.y + b2.y, 0.f);
  m.z = nm * fmaxf(hl.z + b0.z + b1.z + b2.z, 0.f);
  m.w = nm * fmaxf(hl.w + b0.w + b1.w + b2.w, 0.f);

  float* dst = agg + (size_t)c * DIM + d0;
  atomicAdd(dst + 0, m.x);
  atomicAdd(dst + 1, m.y);
  atomicAdd(dst + 2, m.z);
  atomicAdd(dst + 3, m.w);
}

// ---------------------------------------------------------------------------
// Self-loop + per-channel partial sums for BatchNorm:
//   v = agg + relu(h_lin + root)/deg ;  stats[d]+=v, stats[128+d]+=v^2
// block = 128 threads (one per channel), each block walks 64 nodes.
// ---------------------------------------------------------------------------
__global__ __launch_bounds__(128) void self_stats_kernel(
    const float* __restrict__ agg, const float* __restrict__ h_lin,
    const float* __restrict__ root_l, const float* __restrict__ deg,
    float* __restrict__ h, float* __restrict__ stats, int n_nodes,
    int nodes_per_block) {
  const int d = threadIdx.x;
  const float re = root_l[d];
  const int n0 = blockIdx.x * nodes_per_block;
  float s = 0.f, sq = 0.f;
  for (int i = 0; i < nodes_per_block; ++i) {
    int n = n0 + i;
    if (n >= n_nodes) break;
    size_t idx = (size_t)n * DIM + d;
    float v = agg[idx] + fmaxf(h_lin[idx] + re, 0.f) / deg[n];
    h[idx] = v;
    s += v;
    sq += v * v;
  }
  atomicAdd(&stats[d], s);
  atomicAdd(&stats[DIM + d], sq);
}

__global__ __launch_bounds__(128) void bn_finalize_kernel(
    const float* __restrict__ stats, const float* __restrict__ gamma_l,
    const float* __restrict__ beta_l, float* __restrict__ sc, float n_inv) {
  const int d = threadIdx.x;
  float mu = stats[d] * n_inv;
  float var = stats[DIM + d] * n_inv - mu * mu;
  float inv = rsqrtf(var + 1e-5f);
  float scale = gamma_l[d] * inv;
  sc[d] = scale;
  sc[DIM + d] = beta_l[d] - mu * scale;
}

__global__ __launch_bounds__(256) void bn_apply_kernel(
    const float* __restrict__ hin, const float* __restrict__ sc,
    float* __restrict__ hout, int total4, int do_relu) {
  const int i = blockIdx.x * 256 + threadIdx.x;
  if (i >= total4) return;
  const int d0 = (i & (DIM / 4 - 1)) * 4;
  float4 v = ((const float4*)hin)[i];
  float4 s = *(const float4*)(sc + d0);
  float4 b = *(const float4*)(sc + DIM + d0);
  float4 o;
  o.x = fmaf(v.x, s.x, b.x);
  o.y = fmaf(v.y, s.y, b.y);
  o.z = fmaf(v.z, s.z, b.z);
  o.w = fmaf(v.w, s.w, b.w);
  if (do_relu) {
    o.x = fmaxf(o.x, 0.f);
    o.y = fmaxf(o.y, 0.f);
    o.z = fmaxf(o.z, 0.f);
    o.w = fmaxf(o.w, 0.f);
  }
  ((float4*)hout)[i] = o;
}

// ---------------------------------------------------------------------------
extern "C" void kernel_launch(void* const* d_in, const int* in_sizes, int n_in,
                              void* d_out, int out_size, void* d_ws,
                              size_t ws_size, hipStream_t stream) {
  const int* x          = (const int*)d_in[0];
  const int* edge_index = (const int*)d_in[1];
  const int* edge_attr  = (const int*)d_in[2];
  const float* atom_emb = (const float*)d_in[3];
  const float* conv_W   = (const float*)d_in[4];
  const float* conv_b   = (const float*)d_in[5];
  const float* root_emb = (const float*)d_in[6];
  const float* bond_emb = (const float*)d_in[7];
  const float* bn_gamma = (const float*)d_in[8];
  const float* bn_beta  = (const float*)d_in[9];
  float* out = (float*)d_out;

  const int Nn = in_sizes[0] / 9;
  const int Ee = in_sizes[1] / 2;
  const int* row = edge_index;       // source
  const int* colp = edge_index + Ee; // target

  const size_t ND = (size_t)Nn * DIM;
  float* ws    = (float*)d_ws;
  float* h     = ws;
  float* h_lin = h + ND;
  float* agg   = h_lin + ND;
  float* deg   = agg + ND;
  float* dinv  = deg + Nn;
  float* nrm   = dinv + Nn;
  float* stats = nrm + Ee;
  float* sc    = stats + 2 * DIM;

  hipMemsetAsync(deg, 0, Nn * sizeof(float), stream);
  atom_encoder_kernel<<<Nn, 128, 0, stream>>>(x, atom_emb, h, Nn);
  deg_accum_kernel<<<(Ee + 255) / 256, 256, 0, stream>>>(row, deg, Ee);
  deg_finalize_kernel<<<(Nn + 255) / 256, 256, 0, stream>>>(deg, dinv, Nn);
  norm_kernel<<<(Ee + 255) / 256, 256, 0, stream>>>(row, colp, dinv, nrm, Ee);

  const int NPB = 64;
  const int total4 = (int)(ND / 4);
  for (int l = 0; l < NUM_L; ++l) {
    gemm_wmma_kernel<<<(Nn + 15) / 16, 256, 0, stream>>>(
        h, conv_W + (size_t)l * DIM * DIM, conv_b + l * DIM, h_lin, Nn);
    hipMemsetAsync(agg, 0, ND * sizeof(float), stream);
    hipMemsetAsync(stats, 0, 2 * DIM * sizeof(float), stream);
    message_kernel<<<(Ee + 7) / 8, 256, 0, stream>>>(
        row, colp, edge_attr, nrm, h_lin,
        bond_emb + (size_t)l * 3 * BOND_V * DIM, agg, Ee);
    self_stats_kernel<<<(Nn + NPB - 1) / NPB, 128, 0, stream>>>(
        agg, h_lin, root_emb + l * DIM, deg, h, stats, Nn, NPB);
    bn_finalize_kernel<<<1, 128, 0, stream>>>(
        stats, bn_gamma + l * DIM, bn_beta + l * DIM, sc, 1.0f / (float)Nn);
    float* dst = (l == NUM_L - 1) ? out : h;
    bn_apply_kernel<<<(total4 + 255) / 256, 256, 0, stream>>>(
        h, sc, dst, total4, (l < NUM_L - 1) ? 1 : 0);
  }
}